// GNN_37641093382232
// MI455X (gfx1250) — compile-verified
//
#include <hip/hip_runtime.h>
#include <math.h>

typedef __attribute__((ext_vector_type(2))) float v2f;
typedef __attribute__((ext_vector_type(8))) float v8f;

#define D_IN  128
#define D_HID 64
#define D_OUT 16

#define SELU_LAMBDA 1.0507009873554805f
#define SELU_ALPHA  1.6732632423543772f
#define BN_EPS      1e-5f

// ---------------------------------------------------------------------------
// GEMM1: out[N x 64] = x[N x 128] @ W1[128 x 64]   (f32 WMMA 16x16x4)
// Block: 128 threads (4 waves). Each block = 16 rows; each wave = one 16-col tile.
// ---------------------------------------------------------------------------
__global__ void gnn_gemm1(const float* __restrict__ x,
                          const float* __restrict__ W1,
                          float* __restrict__ out, int N) {
    __shared__ float tileA[16 * D_IN];           // 8 KB
    const int row0 = blockIdx.x * 16;
    const int tid  = threadIdx.x;                // 0..127

    // Stage 16x128 A tile into LDS (float4, coalesced), zero-pad OOB rows.
    for (int i = tid; i < (16 * D_IN) / 4; i += 128) {
        const int foff = i * 4;
        const int rr = foff >> 7;                // /128
        const int cc = foff & 127;
        float4 v = make_float4(0.f, 0.f, 0.f, 0.f);
        if (row0 + rr < N)
            v = *(const float4*)(x + (size_t)(row0 + rr) * D_IN + cc);
        ((float4*)tileA)[i] = v;
    }
    __syncthreads();

    const int wave = tid >> 5;                   // 0..3 -> column tile
    const int lane = tid & 31;
    const int half = lane >> 4;                  // 0: lanes 0-15, 1: lanes 16-31
    const int r    = lane & 15;
    const int col0 = wave * 16;

    v8f acc = {};
    #pragma unroll
    for (int k = 0; k < D_IN / 4; ++k) {         // 32 WMMA steps, K=4 each
        const int kk = k * 4 + half * 2;
        v2f a, b;
        a.x = tileA[r * D_IN + kk + 0];          // A[r][kk], A[r][kk+1]
        a.y = tileA[r * D_IN + kk + 1];
        b.x = W1[(kk + 0) * D_HID + col0 + r];   // B[kk][n], n = lane&15
        b.y = W1[(kk + 1) * D_HID + col0 + r];
        acc = __builtin_amdgcn_wmma_f32_16x16x4_f32(
                  false, a, false, b, (short)0, acc, false, false);
    }

    #pragma unroll
    for (int v = 0; v < 8; ++v) {                // D: row = v + 8*half, col = r
        const int row = row0 + v + 8 * half;
        if (row < N) out[(size_t)row * D_HID + col0 + r] = acc[v];
    }
}

// ---------------------------------------------------------------------------
// Edge scatter-add, D=64: out[dst[e]][d] += h[src[e]][d]
// 64 consecutive threads share one edge -> 256B coalesced gather + L2 atomics.
// ---------------------------------------------------------------------------
__global__ void gnn_scatter64(const int* __restrict__ src,
                              const int* __restrict__ dst,
                              const float* __restrict__ h,
                              float* __restrict__ out, long long E) {
    const long long t = (long long)blockIdx.x * blockDim.x + threadIdx.x;
    if (t >= E * 64) return;
    const long long e = t >> 6;
    const int d = (int)(t & 63);
    const int s  = src[e];
    const int dd = dst[e];
    atomicAdd(&out[(size_t)dd * 64 + d], h[(size_t)s * 64 + d]);
}

// Edge scatter-add, D=16
__global__ void gnn_scatter16(const int* __restrict__ src,
                              const int* __restrict__ dst,
                              const float* __restrict__ h,
                              float* __restrict__ out, long long E) {
    const long long t = (long long)blockIdx.x * blockDim.x + threadIdx.x;
    if (t >= E * 16) return;
    const long long e = t >> 4;
    const int d = (int)(t & 15);
    const int s  = src[e];
    const int dd = dst[e];
    atomicAdd(&out[(size_t)dd * 16 + d], h[(size_t)s * 16 + d]);
}

// ---------------------------------------------------------------------------
// BatchNorm statistics: per-channel sum / sumsq over N rows (64 channels).
// ---------------------------------------------------------------------------
__global__ void gnn_bn_stats(const float* __restrict__ h,
                             float* __restrict__ sums, int N) {
    const int c = threadIdx.x & 63;
    float s = 0.f, sq = 0.f;
    const int rowsPerBlock = blockDim.x >> 6;    // 4
    int row = blockIdx.x * rowsPerBlock + (threadIdx.x >> 6);
    const int stride = gridDim.x * rowsPerBlock;
    for (; row < N; row += stride) {
        const float v = h[(size_t)row * 64 + c];
        s += v; sq += v * v;
    }
    atomicAdd(&sums[c], s);
    atomicAdd(&sums[64 + c], sq);
}

// bnp: mu[64], scale[64] (= bn_w * rsqrt(var+eps)), beta[64]
__global__ void gnn_bn_finalize(const float* __restrict__ sums,
                                const float* __restrict__ bn_w,
                                const float* __restrict__ bn_b,
                                float* __restrict__ bnp, int N) {
    const int c = threadIdx.x;
    if (c >= 64) return;
    const float invN = 1.f / (float)N;
    const float mu  = sums[c] * invN;
    const float var = sums[64 + c] * invN - mu * mu;
    bnp[c]        = mu;
    bnp[64 + c]   = bn_w[c] * rsqrtf(var + BN_EPS);
    bnp[128 + c]  = bn_b[c];
}

// ---------------------------------------------------------------------------
// GEMM2 with fused BN-apply + SELU on the A operand:
// out[N x 16] = selu(bn(h[N x 64])) @ W2[64 x 16]
// Block: 128 threads (4 waves); block covers 64 rows; each wave one 16-row tile.
// ---------------------------------------------------------------------------
__global__ void gnn_gemm2(const float* __restrict__ h,
                          const float* __restrict__ W2,
                          const float* __restrict__ bnp,
                          float* __restrict__ out, int N) {
    __shared__ float tile[64 * 64];              // 16 KB
    const int row0 = blockIdx.x * 64;
    const int tid  = threadIdx.x;

    for (int i = tid; i < 64 * 64; i += 128) {
        const int rr = i >> 6;
        const int c  = i & 63;
        const int row = row0 + rr;
        float v = 0.f;
        if (row < N) {
            const float t = (h[(size_t)row * 64 + c] - bnp[c]) * bnp[64 + c]
                            + bnp[128 + c];
            v = (t > 0.f) ? SELU_LAMBDA * t
                          : SELU_LAMBDA * SELU_ALPHA * (expf(t) - 1.f);
        }
        tile[i] = v;
    }
    __syncthreads();

    const int wave = tid >> 5;
    const int lane = tid & 31;
    const int half = lane >> 4;
    const int r    = lane & 15;
    const int wrow0 = wave * 16;
    if (row0 + wrow0 >= N) return;

    v8f acc = {};
    #pragma unroll
    for (int k = 0; k < D_HID / 4; ++k) {        // 16 WMMA steps
        const int kk = k * 4 + half * 2;
        v2f a, b;
        a.x = tile[(wrow0 + r) * 64 + kk + 0];
        a.y = tile[(wrow0 + r) * 64 + kk + 1];
        b.x = W2[(kk + 0) * D_OUT + r];
        b.y = W2[(kk + 1) * D_OUT + r];
        acc = __builtin_amdgcn_wmma_f32_16x16x4_f32(
                  false, a, false, b, (short)0, acc, false, false);
    }

    #pragma unroll
    for (int v = 0; v < 8; ++v) {
        const int row = row0 + wrow0 + v + 8 * half;
        if (row < N) out[(size_t)row * D_OUT + r] = acc[v];
    }
}

// ---------------------------------------------------------------------------
// In-place: z = row + b2 ; out = z - max(z) - log(sum(exp(z - max)))
// ---------------------------------------------------------------------------
__global__ void gnn_logsoftmax(float* __restrict__ inout,
                               const float* __restrict__ b2, int N) {
    const int row = blockIdx.x * blockDim.x + threadIdx.x;
    if (row >= N) return;
    float* p = inout + (size_t)row * D_OUT;
    float v[D_OUT];
    float mx = -INFINITY;
    #pragma unroll
    for (int i = 0; i < D_OUT; ++i) {
        v[i] = p[i] + b2[i];
        mx = fmaxf(mx, v[i]);
    }
    float s = 0.f;
    #pragma unroll
    for (int i = 0; i < D_OUT; ++i) s += expf(v[i] - mx);
    const float ls = logf(s);
    #pragma unroll
    for (int i = 0; i < D_OUT; ++i) p[i] = v[i] - mx - ls;
}

// ---------------------------------------------------------------------------
extern "C" void kernel_launch(void* const* d_in, const int* in_sizes, int n_in,
                              void* d_out, int out_size, void* d_ws, size_t ws_size,
                              hipStream_t stream) {
    const float* x        = (const float*)d_in[0];
    const int*   edge_src = (const int*)d_in[1];
    const int*   edge_dst = (const int*)d_in[2];
    const float* W1       = (const float*)d_in[3];
    // d_in[4] = b1: cancels exactly through training-mode BatchNorm (constant
    // per-channel shift to both h and its batch mean) -> intentionally unused.
    const float* bn_w     = (const float*)d_in[5];
    const float* bn_b     = (const float*)d_in[6];
    const float* W2       = (const float*)d_in[7];
    const float* b2       = (const float*)d_in[8];

    const int N = in_sizes[0] / D_IN;
    const long long E = in_sizes[1];

    // Workspace layout (f32): A[N*64] | B[N*64] | C[N*16] | sums[128] | bnp[192]
    float* A    = (float*)d_ws;
    float* B    = A + (size_t)N * D_HID;
    float* C    = B + (size_t)N * D_HID;
    float* sums = C + (size_t)N * D_OUT;
    float* bnp  = sums + 128;

    float* outp = (float*)d_out;

    const size_t bytes64 = (size_t)N * D_HID * sizeof(float);
    const size_t bytes16 = (size_t)N * D_OUT * sizeof(float);

    // 1) A = x @ W1          (hop/linear commuted: project to 64 dims first)
    gnn_gemm1<<<(N + 15) / 16, 128, 0, stream>>>(x, W1, A, N);

    // 2) B = (Adj + I) A     (copy self term, then scatter-add neighbors)
    hipMemcpyAsync(B, A, bytes64, hipMemcpyDeviceToDevice, stream);
    {
        const long long work = E * 64;
        gnn_scatter64<<<(int)((work + 255) / 256), 256, 0, stream>>>(
            edge_src, edge_dst, A, B, E);
    }

    // 3) A = (Adj + I) B
    hipMemcpyAsync(A, B, bytes64, hipMemcpyDeviceToDevice, stream);
    {
        const long long work = E * 64;
        gnn_scatter64<<<(int)((work + 255) / 256), 256, 0, stream>>>(
            edge_src, edge_dst, B, A, E);
    }

    // 4) BatchNorm batch statistics over A
    hipMemsetAsync(sums, 0, 128 * sizeof(float), stream);
    gnn_bn_stats<<<256, 256, 0, stream>>>(A, sums, N);
    gnn_bn_finalize<<<1, 64, 0, stream>>>(sums, bn_w, bn_b, bnp, N);

    // 5) C = selu(bn(A)) @ W2   (BN + SELU fused into GEMM2's tile load)
    gnn_gemm2<<<(N + 63) / 64, 128, 0, stream>>>(A, W2, bnp, C, N);

    // 6) out = (Adj + I) C      (hop at D=16 after the projection)
    hipMemcpyAsync(outp, C, bytes16, hipMemcpyDeviceToDevice, stream);
    {
        const long long work = E * 16;
        gnn_scatter16<<<(int)((work + 255) / 256), 256, 0, stream>>>(
            edge_src, edge_dst, C, outp, E);
    }

    // 7) out = log_softmax(out + b2)  (b2 applied after the final hop)
    gnn_logsoftmax<<<(N + 255) / 256, 256, 0, stream>>>(outp, b2, N);
}